// AttnFeatTrans1d_54090818126593
// MI455X (gfx1250) — compile-verified
//
#include <hip/hip_runtime.h>

// AMD CDNA5 (gfx1250) implementation. wave32. Uses V_WMMA_F32_16X16X4_F32
// for both f32 GEMMs; adj is synthesized on the fly (never materialized).
// Round 2: v_rcp/v_sqrt raw ops, no divergent branches in the WMMA loop,
// software-pipelined (double-buffered) global loads feeding the WMMAs.

#define Bb 64
#define Nn 64
#define Cc 8
#define Ff 256
#define Kk 2048   // C*F
#define Oo 2048   // OC*OF
#define Mm 4096   // B*N

typedef float v2f __attribute__((ext_vector_type(2)));
typedef float v8f __attribute__((ext_vector_type(8)));

// ---- workspace layout (float offsets), total 21,238,784 floats ~ 85 MB ----
#define OFF_W1SUM 0
#define OFF_W2SUM 256
#define OFF_S1    512
#define OFF_S2    1024                 // 4096
#define OFF_NBW   5120                 // 131072  nbw[b][c][f]
#define OFF_ZX    136192               // 131072  zx[b][k]
#define OFF_NP    267264               // 8388608 nP[b][c][d/2][n][2]
#define OFF_ZNP   8655872              // 8388608 znP[k/2][m][2]
#define OFF_WCP   17044480             // 4194304 WcP[k/2][o][2]

__device__ __forceinline__ v8f wmma4(v2f a, v2f b, v8f c) {
  // D(16x16,f32) = A(16x4,f32) x B(4x16,f32) + C
  return __builtin_amdgcn_wmma_f32_16x16x4_f32(false, a, false, b, (short)0, c,
                                               false, false);
}

// ---------------- column sums of W1/W2 ----------------
__global__ void k_wsums(const float* __restrict__ W1, const float* __restrict__ W2,
                        float* __restrict__ ws) {
  int f = threadIdx.x;
  float a = 0.f, b = 0.f;
#pragma unroll
  for (int k = 0; k < Cc; ++k) { a += W1[k * Ff + f]; b += W2[k * Ff + f]; }
  ws[OFF_W1SUM + f] = a;
  ws[OFF_W2SUM + f] = b;
}

// ---------------- s1[b] = sum x[b]*w1sum ----------------
__global__ void k_s1(const float* __restrict__ x, float* __restrict__ ws) {
  __shared__ float red[256];
  int b = blockIdx.x, t = threadIdx.x;
  float acc = 0.f;
  for (int i = t; i < Kk; i += 256) acc += x[b * Kk + i] * ws[OFF_W1SUM + (i & 255)];
  red[t] = acc; __syncthreads();
  for (int s = 128; s > 0; s >>= 1) { if (t < s) red[t] += red[t + s]; __syncthreads(); }
  if (t == 0) ws[OFF_S1 + b] = red[0];
}

// ---------------- s2[b,n] = sum neighbor[b,n]*w2sum ----------------
__global__ void k_s2(const float* __restrict__ nbr, float* __restrict__ ws) {
  __shared__ float red[256];
  int bn = blockIdx.x, t = threadIdx.x;
  float acc = 0.f;
  for (int i = t; i < Kk; i += 256) acc += nbr[bn * Kk + i] * ws[OFF_W2SUM + (i & 255)];
  red[t] = acc; __syncthreads();
  for (int s = 128; s > 0; s >>= 1) { if (t < s) red[t] += red[t + s]; __syncthreads(); }
  if (t == 0) ws[OFF_S2 + bn] = red[0];
}

// ---------------- nbw[b,c,f] = sum_n (s1*s2) * neighbor ----------------
__global__ void k_nbw(const float* __restrict__ nbr, float* __restrict__ ws) {
  int bc = blockIdx.x, b = bc >> 3, c = bc & 7, f = threadIdx.x;
  float s1 = ws[OFF_S1 + b];
  float acc = 0.f;
  for (int n = 0; n < Nn; ++n) {
    float w = s1 * ws[OFF_S2 + b * Nn + n];
    acc += w * nbr[((b * Nn + n) * Cc + c) * Ff + f];
  }
  ws[OFF_NBW + bc * Ff + f] = acc;
}

// -------- nP[b][c][d/2][n][2] = neighbor[b][n][c][d]  (LDS transpose) --------
__global__ void k_np(const float* __restrict__ nbr, float* __restrict__ ws) {
  __shared__ float tile[64][65];
  int bc = blockIdx.x, b = bc >> 3, c = bc & 7;
  int dbase = blockIdx.y * 64;
  int t = threadIdx.x;
#pragma unroll
  for (int i = 0; i < 16; ++i) {
    int e = i * 256 + t, n = e >> 6, dl = e & 63;
    tile[n][dl] = nbr[((b * Nn + n) * Cc + c) * Ff + dbase + dl];
  }
  __syncthreads();
  float* np = ws + OFF_NP + bc * 16384 + (dbase >> 1) * 128;
#pragma unroll
  for (int i = 0; i < 16; ++i) {
    int e = i * 256 + t, kpl = e >> 7, rem = e & 127;
    int n = rem >> 1, p = rem & 1;
    np[kpl * 128 + rem] = tile[n][2 * kpl + p];
  }
}

// -------- WcP[k/2][o][2] = Wc[o][k]  (LDS transpose) --------
__global__ void k_wcp(const float* __restrict__ Wc, float* __restrict__ ws) {
  __shared__ float tile[32][65];
  int obase = blockIdx.x * 32, kbase = blockIdx.y * 64, t = threadIdx.x;
#pragma unroll
  for (int i = 0; i < 8; ++i) {
    int e = i * 256 + t, ol = e >> 6, kl = e & 63;
    tile[ol][kl] = Wc[(obase + ol) * Kk + kbase + kl];
  }
  __syncthreads();
  v2f* wcp2 = (v2f*)(ws + OFF_WCP);
  int kpbase = kbase >> 1;
#pragma unroll
  for (int i = 0; i < 4; ++i) {
    int e = i * 256 + t, kpl = e >> 5, ol = e & 31;
    v2f st; st.x = tile[ol][2 * kpl]; st.y = tile[ol][2 * kpl + 1];
    wcp2[(kpbase + kpl) * Oo + obase + ol] = st;
  }
}

// ---- fused: adj on-the-fly -> zn = adj x neighbor^T (WMMA), plus zx ----
// grid (16 a-tiles, B), 128 threads = 4 waves, wave w handles n-tile w.
__global__ void __launch_bounds__(128) k_adj_zn(const float* __restrict__ x,
                                                float* __restrict__ ws) {
  __shared__ float lx[Cc * Ff];
  __shared__ float lnb[Cc * Ff];
  int b = blockIdx.y, abase = blockIdx.x * 16, t = threadIdx.x;
  for (int i = t; i < Cc * Ff; i += 128) {
    lx[i] = x[b * Kk + i];
    lnb[i] = ws[OFF_NBW + b * Kk + i];
  }
  __syncthreads();
  int wv = t >> 5, lane = t & 31, lo = lane & 15, hi = lane >> 4;
  int a = abase + lo;
  float xa[Cc], nba[Cc];
#pragma unroll
  for (int c = 0; c < Cc; ++c) { xa[c] = lx[c * Ff + a]; nba[c] = lnb[c * Ff + a]; }
  v8f acc[Cc];
  const v8f z8 = {0.f, 0.f, 0.f, 0.f, 0.f, 0.f, 0.f, 0.f};
#pragma unroll
  for (int c = 0; c < Cc; ++c) acc[c] = z8;
  float zx[Cc];
#pragma unroll
  for (int c = 0; c < Cc; ++c) zx[c] = 0.f;
  const v2f* np2 = (const v2f*)(ws + OFF_NP);
  int nl = wv * 16 + lo;

  // double-buffered B tiles (neighbor^T, pair-packed)
  v2f bvc[Cc], bvn[Cc];
#pragma unroll
  for (int c = 0; c < Cc; ++c)
    bvc[c] = np2[((b * Cc + c) * 128 + hi) * 64 + nl];

  for (int it = 0; it < 64; ++it) {
    int k0 = it * 4;
    int d0 = k0 + 2 * hi, d1 = d0 + 1;
    // prefetch next iteration's B tiles (uniform clamp; long VALU chain hides it)
    int itn = (it + 1 < 64) ? (it + 1) : 63;
#pragma unroll
    for (int c = 0; c < Cc; ++c)
      bvn[c] = np2[((b * Cc + c) * 128 + 2 * itn + hi) * 64 + nl];

    float den0 = 1e-7f, den1 = 1e-7f;
    float s0[Cc], s1v[Cc], xd0a[Cc], xd1a[Cc];
#pragma unroll
    for (int c = 0; c < Cc; ++c) {
      float xd0 = lx[c * Ff + d0], xd1 = lx[c * Ff + d1];
      float t0 = xa[c] * lnb[c * Ff + d0] + xd0 * nba[c];
      float t1 = xa[c] * lnb[c * Ff + d1] + xd1 * nba[c];
      float r0 = __builtin_amdgcn_sqrtf(fmaxf(fabsf(t0), 1e-8f));
      float r1 = __builtin_amdgcn_sqrtf(fmaxf(fabsf(t1), 1e-8f));
      den0 += r0; den1 += r1;
      s0[c] = __builtin_copysignf(r0, t0);
      s1v[c] = __builtin_copysignf(r1, t1);
      xd0a[c] = xd0; xd1a[c] = xd1;
    }
    float rd0 = __builtin_amdgcn_rcpf(den0);
    float rd1 = __builtin_amdgcn_rcpf(den1);
#pragma unroll
    for (int c = 0; c < Cc; ++c) {
      v2f av; av.x = s0[c] * rd0; av.y = s1v[c] * rd1;
      acc[c] = wmma4(av, bvc[c], acc[c]);
    }
    if (wv == 0) {  // single uniform branch per iteration, pure VALU inside
#pragma unroll
      for (int c = 0; c < Cc; ++c)
        zx[c] += (s0[c] * rd0) * xd0a[c] + (s1v[c] * rd1) * xd1a[c];
    }
#pragma unroll
    for (int c = 0; c < Cc; ++c) bvc[c] = bvn[c];
  }
  // store zn K-major pair-packed: znP[k/2][m][2], m = b*64+n
  v2f* znp2 = (v2f*)(ws + OFF_ZNP);
  int m = b * Nn + nl;
#pragma unroll
  for (int c = 0; c < Cc; ++c) {
    int kb = c * Ff + abase + 8 * hi;
#pragma unroll
    for (int v = 0; v < 8; v += 2) {
      v2f st; st.x = acc[c][v]; st.y = acc[c][v + 1];
      znp2[((kb + v) >> 1) * Mm + m] = st;
    }
  }
  if (wv == 0) {
#pragma unroll
    for (int c = 0; c < Cc; ++c) {
      float tot = zx[c] + __shfl_xor(zx[c], 16, 32);
      if (lane < 16) ws[OFF_ZX + b * Kk + c * Ff + a] = tot;
    }
  }
}

// ---- n_out[m][o] = sum_k znP[k][m] * Wc[o][k]  (4096x2048x2048 f32 WMMA) ----
// grid (32, 16), 256 threads = 8 waves; wave w: M-tile16 (mbase), N=128.
__global__ void __launch_bounds__(256) k_gemm_n(const float* __restrict__ ws,
                                                float* __restrict__ out) {
  int t = threadIdx.x;
  int wv = t >> 5, lane = t & 31, lo = lane & 15, hi = lane >> 4;
  int mbase = blockIdx.x * 128 + wv * 16;
  int obase = blockIdx.y * 128;
  const v2f* znp2 = (const v2f*)(ws + OFF_ZNP);
  const v2f* wcp2 = (const v2f*)(ws + OFF_WCP);
  v8f acc[8];
  const v8f z8 = {0.f, 0.f, 0.f, 0.f, 0.f, 0.f, 0.f, 0.f};
#pragma unroll
  for (int j = 0; j < 8; ++j) acc[j] = z8;

  // double-buffered A (zn) + B (WcP) operands
  v2f av_c = znp2[hi * Mm + mbase + lo];
  v2f bv_c[8], bv_n[8];
#pragma unroll
  for (int j = 0; j < 8; ++j) bv_c[j] = wcp2[hi * Oo + obase + j * 16 + lo];

  for (int it = 0; it < 512; ++it) {
    int itn = (it + 1 < 512) ? (it + 1) : 511;
    int kpn = itn * 2 + hi;
    v2f av_n = znp2[kpn * Mm + mbase + lo];
#pragma unroll
    for (int j = 0; j < 8; ++j) bv_n[j] = wcp2[kpn * Oo + obase + j * 16 + lo];
#pragma unroll
    for (int j = 0; j < 8; ++j) acc[j] = wmma4(av_c, bv_c[j], acc[j]);
    av_c = av_n;
#pragma unroll
    for (int j = 0; j < 8; ++j) bv_c[j] = bv_n[j];
  }
  float* nout = out + Bb * Oo;  // after x_out (131072 floats)
#pragma unroll
  for (int j = 0; j < 8; ++j) {
    int o = obase + j * 16 + lo;
#pragma unroll
    for (int v = 0; v < 8; ++v) {
      int m = mbase + v + 8 * hi;
      nout[m * Oo + o] = acc[j][v];
    }
  }
}

// ---- x_out[b][o] = sum_k zx[b][k]*Wc[o][k]  (small GEMV batch) ----
__global__ void k_xout(const float* __restrict__ ws, float* __restrict__ out) {
  __shared__ float zxl[8 * Kk];  // 64 KB
  int t = threadIdx.x, og = blockIdx.x, bg = blockIdx.y;
  for (int i = t; i < 8 * Kk; i += 256)
    zxl[i] = ws[OFF_ZX + (bg * 8 + (i >> 11)) * Kk + (i & 2047)];
  __syncthreads();
  int o = og * 256 + t;
  const v2f* wcp2 = (const v2f*)(ws + OFF_WCP);
  float acc[8];
#pragma unroll
  for (int j = 0; j < 8; ++j) acc[j] = 0.f;
  for (int kp = 0; kp < 1024; ++kp) {
    v2f w = wcp2[kp * Oo + o];
#pragma unroll
    for (int j = 0; j < 8; ++j)
      acc[j] += w.x * zxl[j * Kk + 2 * kp] + w.y * zxl[j * Kk + 2 * kp + 1];
  }
#pragma unroll
  for (int j = 0; j < 8; ++j) out[(bg * 8 + j) * Oo + o] = acc[j];
}

extern "C" void kernel_launch(void* const* d_in, const int* in_sizes, int n_in,
                              void* d_out, int out_size, void* d_ws, size_t ws_size,
                              hipStream_t stream) {
  const float* x   = (const float*)d_in[0];
  const float* nbr = (const float*)d_in[1];
  const float* W1  = (const float*)d_in[2];
  const float* W2  = (const float*)d_in[3];
  const float* Wc  = (const float*)d_in[4];
  float* out = (float*)d_out;
  float* ws  = (float*)d_ws;

  k_wsums<<<1, 256, 0, stream>>>(W1, W2, ws);
  k_s1<<<Bb, 256, 0, stream>>>(x, ws);
  k_s2<<<Bb * Nn, 256, 0, stream>>>(nbr, ws);
  k_nbw<<<Bb * Cc, 256, 0, stream>>>(nbr, ws);
  k_np<<<dim3(Bb * Cc, 4), 256, 0, stream>>>(nbr, ws);
  k_wcp<<<dim3(64, 32), 256, 0, stream>>>(Wc, ws);
  k_adj_zn<<<dim3(16, Bb), 128, 0, stream>>>(x, ws);
  k_gemm_n<<<dim3(32, 16), 256, 0, stream>>>(ws, out);
  k_xout<<<dim3(8, 8), 256, 0, stream>>>(ws, out);
}